// AlternativeIdeaModel_4346506903645
// MI455X (gfx1250) — compile-verified
//
#include <hip/hip_runtime.h>
#include <hip/hip_bf16.h>

typedef float v2f __attribute__((ext_vector_type(2)));
typedef float v8f __attribute__((ext_vector_type(8)));
typedef int   vs4i __attribute__((vector_size(16)));   // async-LDS builtin pointee types
typedef int   vs2i __attribute__((vector_size(8)));

#define S_     5000
#define C_     20000
#define G_ST_  2000
#define G_SC_  2000
#define D_     32
#define K_     20
#define H_ENC_ 64
#define H_DEC_ 256
#define E_     80000

#ifndef __has_builtin
#define __has_builtin(x) 0
#endif

#define AS_GLOBAL __attribute__((address_space(1)))
#define AS_LOCAL  __attribute__((address_space(3)))

#if __has_builtin(__builtin_amdgcn_global_load_async_to_lds_b128) && \
    __has_builtin(__builtin_amdgcn_global_load_async_to_lds_b64)
#define HAVE_ASYNC_LDS 1
#else
#define HAVE_ASYNC_LDS 0
#endif

// ---- global -> LDS copy helpers (async DMA path on CDNA5, register fallback otherwise) ----
__device__ __forceinline__ void cp16_g2l(float* dst_lds, const float* src_g) {
#if HAVE_ASYNC_LDS
    __builtin_amdgcn_global_load_async_to_lds_b128(
        (AS_GLOBAL vs4i*)src_g, (AS_LOCAL vs4i*)dst_lds, 0, 0);
#else
    float4 t = *(const float4*)src_g;
    *(float4*)dst_lds = t;
#endif
}

__device__ __forceinline__ void cp8_g2l(float* dst_lds, const float* src_g) {
#if HAVE_ASYNC_LDS
    __builtin_amdgcn_global_load_async_to_lds_b64(
        (AS_GLOBAL vs2i*)src_g, (AS_LOCAL vs2i*)dst_lds, 0, 0);
#else
    float2 t = *(const float2*)src_g;
    *(float2*)dst_lds = t;
#endif
}

template<int N>
__device__ __forceinline__ void wait_asynccnt_le() {
#if HAVE_ASYNC_LDS
#if __has_builtin(__builtin_amdgcn_s_wait_asynccnt)
    __builtin_amdgcn_s_wait_asynccnt(N);
#else
    asm volatile("s_wait_asynccnt %0" :: "i"(N) : "memory");
#endif
#endif
}

// ---------------- utility kernels ----------------

__global__ void zero_kernel(float* __restrict__ p, int n) {
    int i = blockIdx.x * blockDim.x + threadIdx.x;
    if (i < n) p[i] = 0.0f;
}

__global__ void copy_kernel(const float* __restrict__ src, float* __restrict__ dst, int n) {
    int i = blockIdx.x * blockDim.x + threadIdx.x;
    if (i < n) dst[i] = src[i];
}

__global__ void deg_kernel(const int* __restrict__ ei, float* __restrict__ deg) {
    int e = blockIdx.x * blockDim.x + threadIdx.x;
    if (e < E_) atomicAdd(&deg[ei[E_ + e]], 1.0f);
}

__global__ void dinv_kernel(const float* __restrict__ deg, float* __restrict__ dinv,
                            float* __restrict__ invdeg) {
    int i = blockIdx.x * blockDim.x + threadIdx.x;
    if (i < S_) {
        float d = deg[i] + 1.0f;
        dinv[i] = rsqrtf(d);
        invdeg[i] = 1.0f / d;
    }
}

// ---------------- WMMA fp32 GEMM with double-buffered async LDS staging ----------------
// C[M,N] = A[M,K] @ B[K,N]; N == BN; BK must divide Kdim exactly.
// Block = 256 threads = 8 waves; each wave owns one 16x16 tile of the BMxBN block tile.
template<int BM, int BN, int BK>
__global__ __launch_bounds__(256)
void gemm_wmma_f32(const float* __restrict__ A, const float* __restrict__ B,
                   float* __restrict__ Cout, int M, int Kdim, int N) {
    constexpr int PAD = 4;                         // keep rows 16B-aligned, dodge bank conflicts
    constexpr int UA  = (BK % 4 == 0) ? 4 : 2;     // A copy granularity (floats)
    constexpr int CA  = BM * BK / UA;              // A chunks per tile (128 or 512 in the launches below)
    constexpr int CB  = BK * BN / 4;               // B chunks per tile (multiple of 256)
    constexpr int NOPS = CA / 256 + CB / 256;      // async ops issued per thread per tile

    __shared__ float As[2][BM][BK + PAD];
    __shared__ float Bs[2][BK][BN];

    const int tid   = threadIdx.x;
    const int lane  = tid & 31;
    const int wave  = tid >> 5;
    const int wm    = wave / (BN / 16);
    const int wn    = wave % (BN / 16);
    const int row0  = blockIdx.x * BM;
    const int khalf = lane >> 4;
    const int l16   = lane & 15;

    auto issue = [&](int sel, int kb) {
        #pragma unroll
        for (int i = 0; i < CA / 256; ++i) {
            int e = tid + i * 256;
            int r = e / (BK / UA);
            int c = (e % (BK / UA)) * UA;
            int gr = row0 + r;
            if (gr >= M) gr = M - 1;               // clamp: junk rows are never stored
            const float* src = &A[(size_t)gr * Kdim + kb + c];
            if constexpr (UA == 4) cp16_g2l(&As[sel][r][c], src);
            else                   cp8_g2l(&As[sel][r][c], src);
        }
        #pragma unroll
        for (int i = 0; i < CB / 256; ++i) {
            int e = tid + i * 256;
            int r = e / (BN / 4);
            int c = (e % (BN / 4)) * 4;
            cp16_g2l(&Bs[sel][r][c], &B[(size_t)(kb + r) * N + c]);
        }
    };

    v8f acc = {};
    const int nch = Kdim / BK;

    issue(0, 0);
    for (int t = 0; t < nch; ++t) {
        const int cur = t & 1;
        const bool hasnext = (t + 1 < nch);
        if (hasnext) {
            issue(cur ^ 1, (t + 1) * BK);
            wait_asynccnt_le<NOPS>();              // current tile's DMA done; next in flight
        } else {
            wait_asynccnt_le<0>();
        }
        __syncthreads();

        #pragma unroll
        for (int kk = 0; kk < BK; kk += 4) {
            v2f av, bv;
            av.x = As[cur][wm * 16 + l16][kk + 2 * khalf + 0];
            av.y = As[cur][wm * 16 + l16][kk + 2 * khalf + 1];
            bv.x = Bs[cur][kk + 2 * khalf + 0][wn * 16 + l16];
            bv.y = Bs[cur][kk + 2 * khalf + 1][wn * 16 + l16];
            acc = __builtin_amdgcn_wmma_f32_16x16x4_f32(
                false, av, false, bv, (short)0, acc, false, false);
        }
        __syncthreads();
    }

    #pragma unroll
    for (int j = 0; j < 8; ++j) {
        int r = row0 + wm * 16 + j + 8 * khalf;
        int c = wn * 16 + l16;
        if (r < M) Cout[(size_t)r * N + c] = acc[j];
    }
}

// ---------------- GCN edge scatter ----------------
__global__ void scatter_kernel(const float* __restrict__ xw, float* __restrict__ agg,
                               const int* __restrict__ ei, const float* __restrict__ dinv,
                               int Dd) {
    int gid = blockIdx.x * blockDim.x + threadIdx.x;
    if (gid >= E_ * Dd) return;
    int e = gid / Dd;
    int d = gid % Dd;
    int r = ei[e];
    int c = ei[E_ + e];
    float coef = dinv[r] * dinv[c];
    atomicAdd(&agg[(size_t)c * Dd + d], xw[(size_t)r * Dd + d] * coef);
}

__global__ void finish_kernel(const float* __restrict__ agg, const float* __restrict__ xw,
                              const float* __restrict__ invdeg, const float* __restrict__ bias,
                              float* __restrict__ out, int Dd, int n, int do_relu) {
    int i = blockIdx.x * blockDim.x + threadIdx.x;
    if (i >= n) return;
    int row = i / Dd, col = i % Dd;
    float v = agg[i] + xw[i] * invdeg[row] + bias[col];
    out[i] = do_relu ? fmaxf(v, 0.0f) : v;
}

// ---------------- row softmax of U [S_, C_]: async double-buffered row pipeline ----------------
// Each block walks rows with stride SM_BLOCKS; while reducing row r it DMAs row r+stride.
#define SM_BLOCKS 640
#define ROWPAD    20480    // 5120 16B-chunks per row buffer (5000 real + clamp padding)

__global__ __launch_bounds__(1024)
void softmax_rows(const float* __restrict__ U, float* __restrict__ Aout) {
    __shared__ float buf[2][ROWPAD];   // 160 KB double buffer (320 KB LDS / WGP on CDNA5)
    __shared__ float redm[32];
    __shared__ float reds[32];
    const int tid = threadIdx.x;

    auto issue_row = [&](int sel, int row) {
        const float* src = U + (size_t)row * C_;
        #pragma unroll
        for (int i = 0; i < 5; ++i) {              // 5 ops/thread, uniform per wave
            int c = tid + i * 1024;                // 16B chunk index 0..5119
            int gc = (c < 5000) ? c : 4999;        // clamp source; pad slots get dup data
            cp16_g2l(&buf[sel][4 * c], src + 4 * gc);
        }
    };

    int row = blockIdx.x;
    issue_row(0, row);
    for (int t = 0; row < S_; row += SM_BLOCKS, ++t) {
        const int cur = t & 1;
        const int nrow = row + SM_BLOCKS;
        if (nrow < S_) {
            issue_row(cur ^ 1, nrow);
            wait_asynccnt_le<5>();                 // current row landed; next row in flight
        } else {
            wait_asynccnt_le<0>();
        }
        __syncthreads();

        float lmax = -3.4e38f;
        for (int j = tid; j < C_; j += 1024) lmax = fmaxf(lmax, buf[cur][j]);
        #pragma unroll
        for (int off = 16; off; off >>= 1) lmax = fmaxf(lmax, __shfl_xor(lmax, off, 32));
        if ((tid & 31) == 0) redm[tid >> 5] = lmax;
        __syncthreads();
        float m = redm[tid & 31];                  // 32 waves -> all slots valid
        #pragma unroll
        for (int off = 16; off; off >>= 1) m = fmaxf(m, __shfl_xor(m, off, 32));

        float lsum = 0.0f;
        for (int j = tid; j < C_; j += 1024) {
            float e = __expf(buf[cur][j] - m);
            buf[cur][j] = e;
            lsum += e;
        }
        #pragma unroll
        for (int off = 16; off; off >>= 1) lsum += __shfl_xor(lsum, off, 32);
        if ((tid & 31) == 0) reds[tid >> 5] = lsum;
        __syncthreads();
        float s = reds[tid & 31];
        #pragma unroll
        for (int off = 16; off; off >>= 1) s += __shfl_xor(s, off, 32);

        float inv = 1.0f / s;
        const size_t base = (size_t)row * C_;
        for (int j = tid; j < C_; j += 1024) Aout[base + j] = buf[cur][j] * inv;
        __syncthreads();                           // all readers done before buffer reuse
    }
}

// ---------------- softmax of V [C_,K_] + column sums psum[k] = sum_c B[c,k] ----------------
__global__ __launch_bounds__(256)
void softmax_V_kernel(const float* __restrict__ V, float* __restrict__ B,
                      float* __restrict__ psum) {
    __shared__ float sp[K_];
    const int tid = threadIdx.x;
    if (tid < K_) sp[tid] = 0.0f;
    __syncthreads();
    int r = blockIdx.x * 256 + tid;
    if (r < C_) {
        float v[K_];
        float m = -3.4e38f;
        #pragma unroll
        for (int k = 0; k < K_; ++k) { v[k] = V[(size_t)r * K_ + k]; m = fmaxf(m, v[k]); }
        float s = 0.0f;
        #pragma unroll
        for (int k = 0; k < K_; ++k) { v[k] = __expf(v[k] - m); s += v[k]; }
        float inv = 1.0f / s;
        #pragma unroll
        for (int k = 0; k < K_; ++k) {
            float b = v[k] * inv;
            B[(size_t)r * K_ + k] = b;
            atomicAdd(&sp[k], b);                  // ds_add_f32
        }
    }
    __syncthreads();
    if (tid < K_) atomicAdd(&psum[tid], sp[tid]);
}

// ---------------- decoder ----------------
__global__ __launch_bounds__(256)
void decoder_T_kernel(const float* __restrict__ F, const float* __restrict__ Wd1,
                      const float* __restrict__ bd1, float* __restrict__ T) {
    int k = blockIdx.x, h = threadIdx.x;
    float s = bd1[h];
    #pragma unroll
    for (int d = 0; d < D_; ++d) s += F[k * D_ + d] * Wd1[d * H_DEC_ + h];
    T[k * H_DEC_ + h] = fmaxf(s, 0.0f);
}

__global__ __launch_bounds__(256)
void decoder_M_kernel(const float* __restrict__ T, const float* __restrict__ Wd2,
                      const float* __restrict__ bd2, const float* __restrict__ psum,
                      float* __restrict__ M) {
    int gid = blockIdx.x * 256 + threadIdx.x;
    if (gid >= K_ * G_SC_) return;
    int k = gid / G_SC_, g = gid % G_SC_;
    float s = bd2[g];
    for (int h = 0; h < H_DEC_; ++h) s += T[k * H_DEC_ + h] * Wd2[h * G_SC_ + g];
    M[gid] = s * psum[k];                          // C*p[k] == colsum(B)[k]
}

// ---------------- launch ----------------
extern "C" void kernel_launch(void* const* d_in, const int* in_sizes, int n_in,
                              void* d_out, int out_size, void* d_ws, size_t ws_size,
                              hipStream_t stream) {
    const float* z   = (const float*)d_in[0];
    const int*   ei  = (const int*)  d_in[1];
    const float* W1  = (const float*)d_in[2];
    const float* b1  = (const float*)d_in[3];
    const float* W2  = (const float*)d_in[4];
    const float* b2  = (const float*)d_in[5];
    const float* U   = (const float*)d_in[6];
    const float* V   = (const float*)d_in[7];
    const float* F   = (const float*)d_in[8];
    const float* Wd1 = (const float*)d_in[9];
    const float* bd1 = (const float*)d_in[10];
    const float* Wd2 = (const float*)d_in[11];
    const float* bd2 = (const float*)d_in[12];
    float* out = (float*)d_out;
    float* ws  = (float*)d_ws;

    float* deg    = ws;
    float* dinv   = deg + S_;
    float* invdeg = dinv + S_;
    float* xw1    = invdeg + S_;
    float* agg1   = xw1  + (size_t)S_ * H_ENC_;
    float* h1     = agg1 + (size_t)S_ * H_ENC_;
    float* xw2    = h1   + (size_t)S_ * H_ENC_;
    float* agg2   = xw2  + (size_t)S_ * D_;
    float* psum   = agg2 + (size_t)S_ * D_;
    float* T      = psum + 32;

    float* outA = out;
    float* outB = outA + (size_t)S_ * C_;
    float* outH = outB + (size_t)C_ * K_;
    float* outM = outH + (size_t)S_ * D_;
    float* outF = outM + (size_t)K_ * G_SC_;

    auto blocks = [](int n, int b) { return (n + b - 1) / b; };

    zero_kernel<<<blocks(S_, 256), 256, 0, stream>>>(deg, S_);
    zero_kernel<<<blocks(S_ * H_ENC_, 256), 256, 0, stream>>>(agg1, S_ * H_ENC_);
    zero_kernel<<<blocks(S_ * D_, 256), 256, 0, stream>>>(agg2, S_ * D_);
    zero_kernel<<<1, 256, 0, stream>>>(psum, K_);

    deg_kernel<<<blocks(E_, 256), 256, 0, stream>>>(ei, deg);
    dinv_kernel<<<blocks(S_, 256), 256, 0, stream>>>(deg, dinv, invdeg);

    // layer 1: xw1 = z @ W1  (WMMA f32, BK=16 divides K=2000)
    gemm_wmma_f32<32, 64, 16><<<blocks(S_, 32), 256, 0, stream>>>(z, W1, xw1, S_, G_ST_, H_ENC_);
    scatter_kernel<<<blocks(E_ * H_ENC_, 256), 256, 0, stream>>>(xw1, agg1, ei, dinv, H_ENC_);
    finish_kernel<<<blocks(S_ * H_ENC_, 256), 256, 0, stream>>>(agg1, xw1, invdeg, b1, h1,
                                                                H_ENC_, S_ * H_ENC_, 1);

    // layer 2: xw2 = h1 @ W2  (WMMA f32, BK=32 divides K=64)
    gemm_wmma_f32<64, 32, 32><<<blocks(S_, 64), 256, 0, stream>>>(h1, W2, xw2, S_, H_ENC_, D_);
    scatter_kernel<<<blocks(E_ * D_, 256), 256, 0, stream>>>(xw2, agg2, ei, dinv, D_);
    finish_kernel<<<blocks(S_ * D_, 256), 256, 0, stream>>>(agg2, xw2, invdeg, b2, outH,
                                                            D_, S_ * D_, 0);

    // A = softmax(U): bandwidth-dominant; async DMA double-buffered row pipeline
    softmax_rows<<<SM_BLOCKS, 1024, 0, stream>>>(U, outA);

    softmax_V_kernel<<<blocks(C_, 256), 256, 0, stream>>>(V, outB, psum);

    decoder_T_kernel<<<K_, 256, 0, stream>>>(F, Wd1, bd1, T);
    decoder_M_kernel<<<blocks(K_ * G_SC_, 256), 256, 0, stream>>>(T, Wd2, bd2, psum, outM);

    copy_kernel<<<blocks(K_ * D_, 256), 256, 0, stream>>>(F, outF, K_ * D_);
}